// FastLIF_69277822485150
// MI455X (gfx1250) — compile-verified
//
#include <hip/hip_runtime.h>
#include <stdint.h>

// FastLIF scan for MI455X (gfx1250, wave32).
// Bandwidth-bound: 1 GiB traffic -> ~46 us roofline @ 23.3 TB/s.
// Uses the CDNA5 async copy engine (GLOBAL_LOAD_ASYNC_TO_LDS_B128 + ASYNCcnt)
// for a 16-deep software pipeline of timestep tiles staged in LDS.

typedef float v4f __attribute__((ext_vector_type(4)));

constexpr int kT      = 512;                 // timesteps (sequential scan)
constexpr int kLanes  = 32 * 64 * 128;       // floats per timestep slab
constexpr int kLanes4 = kLanes / 4;          // 65536 float4 lanes
constexpr int kBlock  = 256;                 // 8 waves of 32
constexpr int kGrid   = kLanes4 / kBlock;    // 256 blocks
constexpr int kDepth  = 16;                  // async pipeline depth -> 64 KB LDS/block

constexpr float kDecay     = 0.9f;
constexpr float kThreshold = 1.0f;

__device__ __forceinline__ v4f lif_step(v4f& s, v4f xv) {
    v4f z;
#pragma unroll
    for (int i = 0; i < 4; ++i) {
        float v = __fmaf_rn(kDecay, s[i], xv[i]);   // v = decay*s + x
        bool spike = v > kThreshold;                // z = H(v - 1)
        z[i] = spike ? 1.0f : 0.0f;
        s[i] = spike ? 0.0f : v;                    // reset-to-zero
    }
    return z;
}

__global__ __launch_bounds__(kBlock) void FastLIF_kernel(
        const float* __restrict__ xg, float* __restrict__ zg) {
    __shared__ v4f stage[kDepth][kBlock];           // 64 KB circular staging buffer

    const int tid   = threadIdx.x;
    const int lane4 = blockIdx.x * kBlock + tid;    // 0 .. kLanes4-1

    const v4f* __restrict__ x4 = reinterpret_cast<const v4f*>(xg) + lane4;
    v4f*       __restrict__ z4 = reinterpret_cast<v4f*>(zg)       + lane4;

    // Low 32 bits of the generic pointer to an LDS object == LDS byte offset.
    const unsigned lds0 = (unsigned)(uintptr_t)&stage[0][tid];

    // ---- prologue: put kDepth timestep tiles in flight on the async engine ----
#pragma unroll
    for (int d = 0; d < kDepth; ++d) {
        const v4f* g  = x4 + (size_t)d * kLanes4;
        unsigned   lo = lds0 + (unsigned)(d * kBlock * sizeof(v4f));
        asm volatile("global_load_async_to_lds_b128 %0, %1, off"
                     :: "v"(lo), "v"(g)
                     : "memory");
    }

    v4f s = {0.0f, 0.0f, 0.0f, 0.0f};
    const v4f* gnext = x4 + (size_t)kDepth * kLanes4;   // next slab to prefetch
    v4f*       gout  = z4;

    // ---- steady state: wait oldest tile, consume, store spikes, refill slot ----
    for (int t = 0; t < kT - kDepth; ++t) {
        // In-order async completion: <= kDepth-1 outstanding => oldest tile landed.
        asm volatile("s_wait_asynccnt %0" :: "i"(kDepth - 1) : "memory");

        v4f xv = stage[t & (kDepth - 1)][tid];          // ds_load_b128
        v4f zs = lif_step(s, xv);

        __builtin_nontemporal_store(zs, gout);          // streamed once, NT store
        gout += kLanes4;

        // Refill the slot just consumed. Dummy dep on s.x (derived from xv)
        // forces issue after the ds_load's s_wait_dscnt => no LDS WAR hazard.
        unsigned lo = lds0 + (unsigned)((t & (kDepth - 1)) * kBlock * sizeof(v4f));
        asm volatile("global_load_async_to_lds_b128 %0, %1, off"
                     :: "v"(lo), "v"(gnext), "v"(s.x)
                     : "memory");
        gnext += kLanes4;
    }

    // ---- drain: everything already in LDS ----
    asm volatile("s_wait_asynccnt 0" ::: "memory");
#pragma unroll
    for (int t = kT - kDepth; t < kT; ++t) {
        v4f xv = stage[t & (kDepth - 1)][tid];
        v4f zs = lif_step(s, xv);
        __builtin_nontemporal_store(zs, gout);
        gout += kLanes4;
    }
}

extern "C" void kernel_launch(void* const* d_in, const int* in_sizes, int n_in,
                              void* d_out, int out_size, void* d_ws, size_t ws_size,
                              hipStream_t stream) {
    (void)in_sizes; (void)n_in; (void)out_size; (void)d_ws; (void)ws_size;
    const float* x = (const float*)d_in[0];
    float*       z = (float*)d_out;
    FastLIF_kernel<<<kGrid, kBlock, 0, stream>>>(x, z);
}